// Ranking_Loss_51496657879409
// MI455X (gfx1250) — compile-verified
//
#include <hip/hip_runtime.h>
#include <hip/hip_bf16.h>
#include <math.h>

// ---------------- configuration ----------------
#define NGROUPS 128
#define CHUNK   8192
#define BLK1    256
#define BLK3    128
#define ITEMS3  (CHUNK / BLK3)   // 64 items per thread in rank pass
#define BLK4    256
#define NBLK4   2048             // must be a multiple of 64 for the WMMA reduce
#define ONE_P_THETA 1.02f

typedef float v2f __attribute__((ext_vector_type(2)));
typedef float v8f __attribute__((ext_vector_type(8)));

__device__ __forceinline__ float sigmoidf(float x) {
    return 1.0f / (1.0f + __expf(-x));
}
// numerically stable softplus(x) = log(1+exp(x))
__device__ __forceinline__ float softplusf(float x) {
    return log1pf(__expf(-fabsf(x))) + fmaxf(x, 0.0f);
}

// ---------------- pass 1: per-chunk group histograms ----------------
__global__ void k_hist(const int* __restrict__ uid, int n, int* __restrict__ chunkHist) {
    __shared__ int h[NGROUPS];
    for (int i = threadIdx.x; i < NGROUPS; i += blockDim.x) h[i] = 0;
    __syncthreads();
    const int start = blockIdx.x * CHUNK;
    for (int o = threadIdx.x; o < CHUNK; o += blockDim.x) {
        int idx = start + o;
        if (idx < n) atomicAdd(&h[uid[idx] & (NGROUPS - 1)], 1);
    }
    __syncthreads();
    for (int i = threadIdx.x; i < NGROUPS; i += blockDim.x)
        chunkHist[blockIdx.x * NGROUPS + i] = h[i];
}

// ---------------- pass 2: cross-chunk prefix + group offsets ----------------
__global__ void k_scan(const int* __restrict__ chunkHist, int nch,
                       int* __restrict__ chunkBase, int* __restrict__ gOff,
                       int* __restrict__ gCnt) {
    __shared__ int counts[NGROUPS];
    const int g = threadIdx.x;           // 128 threads
    int run = 0;
    for (int b = 0; b < nch; b++) {
        int v = chunkHist[b * NGROUPS + g];
        chunkBase[b * NGROUPS + g] = run;
        run += v;
    }
    counts[g] = run;
    gCnt[g] = run;
    __syncthreads();
    if (g == 0) {
        int acc = 0;
        for (int i = 0; i < NGROUPS; i++) { gOff[i] = acc; acc += counts[i]; }
    }
}

// ---------------- pass 3: build rank -> original-index table ----------------
// Deterministic: each thread owns a private LDS counter row; scan order is a
// fixed (thread, item) lexicographic order.
__global__ void k_rank(const int* __restrict__ uid, int n,
                       const int* __restrict__ chunkBase,
                       const int* __restrict__ gOff,
                       int* __restrict__ rankIdx) {
    __shared__ int cnt[BLK3 * NGROUPS];   // 64 KB
    __shared__ int soff[NGROUPS];
    const int t = threadIdx.x;
    const int b = blockIdx.x;
    const int base = b * CHUNK;

    for (int i = 0; i < NGROUPS; i++) cnt[t * NGROUPS + i] = 0;
    soff[t] = gOff[t];                    // BLK3 == NGROUPS == 128
    __syncthreads();

    // phase A: per-thread histogram (coalesced strided item assignment)
    for (int u = 0; u < ITEMS3; u++) {
        int idx = base + u * BLK3 + t;
        if (idx < n) cnt[t * NGROUPS + (uid[idx] & (NGROUPS - 1))]++;
    }
    __syncthreads();

    // phase B: per-bin exclusive scan across threads, seeded with cross-chunk base
    {
        int run = chunkBase[b * NGROUPS + t];   // thread t handles bin t
        for (int t2 = 0; t2 < BLK3; t2++) {
            int tmp = cnt[t2 * NGROUPS + t];
            cnt[t2 * NGROUPS + t] = run;
            run += tmp;
        }
    }
    __syncthreads();

    // phase C: emit rank -> index (each thread touches only its own LDS row)
    for (int u = 0; u < ITEMS3; u++) {
        int idx = base + u * BLK3 + t;
        if (idx < n) {
            int g = uid[idx] & (NGROUPS - 1);
            int r = cnt[t * NGROUPS + g]++;
            rankIdx[soff[g] + r] = idx;
        }
    }
}

// ---------------- pass 4: main loss computation ----------------
__global__ void k_loss(const float* __restrict__ gt, const float* __restrict__ pred,
                       const int* __restrict__ rankIdx, const int* __restrict__ gOff,
                       int n,
                       float* __restrict__ pLog, float* __restrict__ pSq,
                       float* __restrict__ pCnt) {
    __shared__ int soff[NGROUPS + 1];
    __shared__ float red[BLK4];
    for (int i = threadIdx.x; i < NGROUPS; i += BLK4) soff[i] = gOff[i];
    if (threadIdx.x == 0) soff[NGROUPS] = n;
    __syncthreads();

    float accLog = 0.0f, accSq = 0.0f, accCnt = 0.0f;
    const int stride = NBLK4 * BLK4;
    for (int k = blockIdx.x * BLK4 + threadIdx.x; k < n; k += stride) {
        if (k + stride < n) __builtin_prefetch(&rankIdx[k + stride], 0, 0); // global_prefetch_b8

        // binary search: group g with soff[g] <= k < soff[g+1]
        int lo = 0, hi = NGROUPS;
        while (hi - lo > 1) { int mid = (lo + hi) >> 1; if (soff[mid] <= k) lo = mid; else hi = mid; }
        const int g = lo;
        const int off = soff[g];
        const int s = soff[g + 1] - off;
        const int r = k - off;

        // deterministic pseudo-random within-group bijection (cycle-walked LCG)
        unsigned rp = (unsigned)r;
        if (s > 1) {
            unsigned m = (2u << (31 - __clz(s - 1))) - 1u;      // next_pow2(s) - 1
            const unsigned a = 0x9E3779B5u;                     // a % 4 == 1 (full period)
            const unsigned c = ((unsigned)g * 2654435761u) | 1u;
            unsigned x = (unsigned)r;
            do { x = (a * x + c) & m; } while (x >= (unsigned)s);
            rp = x;
        }

        const int i = rankIdx[k];
        const int j = rankIdx[off + (int)rp];   // L2-resident gather (192MB L2 holds all)

        const float za = gt[i], zb = gt[j];
        const float pa = pred[i], pb = pred[j];
        const float tgt = (za > ONE_P_THETA * zb) ? 1.0f
                        : ((zb > ONE_P_THETA * za) ? -1.0f : 0.0f);
        const float d = sigmoidf(pa) - sigmoidf(pb);
        if (tgt != 0.0f) { accLog += softplusf(-tgt * d); accCnt += 1.0f; }
        else             { accSq  += d * d; }
    }

    // fixed-order block tree reductions (deterministic)
    red[threadIdx.x] = accLog; __syncthreads();
    for (int st = BLK4 / 2; st > 0; st >>= 1) {
        if (threadIdx.x < st) red[threadIdx.x] += red[threadIdx.x + st];
        __syncthreads();
    }
    if (threadIdx.x == 0) pLog[blockIdx.x] = red[0];
    __syncthreads();

    red[threadIdx.x] = accSq; __syncthreads();
    for (int st = BLK4 / 2; st > 0; st >>= 1) {
        if (threadIdx.x < st) red[threadIdx.x] += red[threadIdx.x + st];
        __syncthreads();
    }
    if (threadIdx.x == 0) pSq[blockIdx.x] = red[0];
    __syncthreads();

    red[threadIdx.x] = accCnt; __syncthreads();
    for (int st = BLK4 / 2; st > 0; st >>= 1) {
        if (threadIdx.x < st) red[threadIdx.x] += red[threadIdx.x + st];
        __syncthreads();
    }
    if (threadIdx.x == 0) pCnt[blockIdx.x] = red[0];
}

// ---------------- pass 5: single-wave WMMA reduction of block partials ----------------
// Sum via D = A x ones(4x16) + C : sum of ALL D entries == 16 * sum of all A
// entries regardless of VGPR layout, so we stuff 64 partials per step into A.
__global__ void k_final(const float* __restrict__ pLog, const float* __restrict__ pSq,
                        const float* __restrict__ pCnt, int n, float* __restrict__ out) {
    __shared__ float red[32];
    const int lane = threadIdx.x;          // exactly 32 threads, EXEC all-ones
    const v2f ones = {1.0f, 1.0f};
    float totals[3];
    const float* __restrict__ arrs[3] = {pLog, pSq, pCnt};

#pragma unroll
    for (int q = 0; q < 3; q++) {
        const float* __restrict__ p = arrs[q];
        v8f c = {};
        for (int b = 0; b < NBLK4; b += 64) {
            v2f a;
            a.x = p[b + lane];
            a.y = p[b + 32 + lane];
            c = __builtin_amdgcn_wmma_f32_16x16x4_f32(
                    false, a, false, ones, (short)0, c, false, false);
        }
        float s = c[0] + c[1] + c[2] + c[3] + c[4] + c[5] + c[6] + c[7];
        red[lane] = s;
        __syncthreads();
        if (lane == 0) {
            float tot = 0.0f;
            for (int i = 0; i < 32; i++) tot += red[i];
            red[0] = tot * (1.0f / 16.0f);   // each A value counted 16x in D
        }
        __syncthreads();
        totals[q] = red[0];
        __syncthreads();
    }

    if (lane == 0) {
        float cnt   = totals[2];
        float cntNz = fmaxf(cnt, 1.0f);
        float cntZ  = fmaxf((float)n - cnt, 1.0f);
        out[0] = totals[0] / cntNz + totals[1] / cntZ;
    }
}

// ---------------- launcher ----------------
extern "C" void kernel_launch(void* const* d_in, const int* in_sizes, int n_in,
                              void* d_out, int out_size, void* d_ws, size_t ws_size,
                              hipStream_t stream) {
    const float* gt   = (const float*)d_in[0];
    const float* pred = (const float*)d_in[1];
    const int*   uid  = (const int*)d_in[2];
    const int n   = in_sizes[0];
    const int nch = (n + CHUNK - 1) / CHUNK;

    char* ws = (char*)d_ws;
    int*   chunkHist = (int*)ws;   ws += (size_t)nch * NGROUPS * sizeof(int);
    int*   chunkBase = (int*)ws;   ws += (size_t)nch * NGROUPS * sizeof(int);
    int*   gOff      = (int*)ws;   ws += (size_t)NGROUPS * sizeof(int);
    int*   gCnt      = (int*)ws;   ws += (size_t)NGROUPS * sizeof(int);
    int*   rankIdx   = (int*)ws;   ws += (size_t)n * sizeof(int);
    float* pLog      = (float*)ws; ws += (size_t)NBLK4 * sizeof(float);
    float* pSq       = (float*)ws; ws += (size_t)NBLK4 * sizeof(float);
    float* pCnt      = (float*)ws; ws += (size_t)NBLK4 * sizeof(float);

    k_hist <<<nch,   BLK1,    0, stream>>>(uid, n, chunkHist);
    k_scan <<<1,     NGROUPS, 0, stream>>>(chunkHist, nch, chunkBase, gOff, gCnt);
    k_rank <<<nch,   BLK3,    0, stream>>>(uid, n, chunkBase, gOff, rankIdx);
    k_loss <<<NBLK4, BLK4,    0, stream>>>(gt, pred, rankIdx, gOff, n, pLog, pSq, pCnt);
    k_final<<<1,     32,      0, stream>>>(pLog, pSq, pCnt, n, (float*)d_out);
}